// GraphSAGE_66185446031814
// MI455X (gfx1250) — compile-verified
//
#include <hip/hip_runtime.h>

// GraphSAGE on gfx1250 (MI455X), wave32.
// out = act( h@w_self + (segsum(h[src]->dst)*deg_inv)@w_neigh + b ), 3 layers.
// Dense parts use V_WMMA_F32_16X16X4_F32 (full f32 matrix pipe).

typedef __attribute__((ext_vector_type(2))) float v2f;
typedef __attribute__((ext_vector_type(8))) float v8f;

#define NNODES 100000
#define NEDGES 1000000
#define DFEAT  128   // input feature width of every layer (128,128,128)

// ---------------------------------------------------------------- utilities
__global__ void zero4_kernel(float4* __restrict__ p, long n4) {
  long i = (long)blockIdx.x * blockDim.x + threadIdx.x;
  long stride = (long)gridDim.x * blockDim.x;
  const float4 z = make_float4(0.f, 0.f, 0.f, 0.f);
  for (; i < n4; i += stride) p[i] = z;
}

__global__ void degree_kernel(const int* __restrict__ dst, float* __restrict__ deg, int e) {
  int i = blockIdx.x * blockDim.x + threadIdx.x;
  if (i < e) atomicAdd(&deg[dst[i]], 1.0f);
}

__global__ void deginv_kernel(float* __restrict__ deg, int n) {
  int i = blockIdx.x * blockDim.x + threadIdx.x;
  if (i < n) { float d = deg[i]; deg[i] = (d > 0.0f) ? (1.0f / d) : 0.0f; }
}

// one wave (32 lanes) per edge; each lane moves 4 consecutive floats (float4 gather,
// 4x global_atomic_add_f32 no-return scatter). Feature matrix (51MB) is L2-resident.
__global__ void scatter_kernel(const float* __restrict__ h,
                               const int* __restrict__ src,
                               const int* __restrict__ dst,
                               float* __restrict__ acc, int e) {
  int wave = (blockIdx.x * blockDim.x + threadIdx.x) >> 5;
  int lane = threadIdx.x & 31;
  if (wave >= e) return;
  int s = src[wave];
  int d = dst[wave];
  const float4 v = *(const float4*)(h + (long)s * DFEAT + lane * 4);
  float* o = acc + (long)d * DFEAT + lane * 4;
  atomicAdd(o + 0, v.x);
  atomicAdd(o + 1, v.y);
  atomicAdd(o + 2, v.z);
  atomicAdd(o + 3, v.w);
}

// ---------------------------------------------------------------- fused SAGE GEMM
// Block = 256 threads = 8 waves; each wave owns a 16-row tile -> 128 rows/block.
// Each wave computes a 16 x NOUT stripe with NOUT/16 f32 accumulators.
// Two K=128 phases share accumulators: phase0 A=h,B=w_self; phase1 A=neigh*dinv,B=w_neigh.
//
// WMMA f32 16x16x4 fragment layouts (ISA 7.12.2):
//  A (16x4): lane<16 holds row M=lane, {K=k,k+1}; lane>=16 row M=lane-16, {K=k+2,k+3}
//  B (4x16): lane<16 holds col N=lane, {K=k,k+1}; lane>=16 col N=lane-16, {K=k+2,k+3}
//  C/D: VGPR j = row (j + 8*(lane>=16)), col = lane%16
//
// LDS weight layout (conflict-free b64 fragment reads, no repacking moves):
//  element W[kk][n] -> lds dword ((kk>>2)*NOUT + n)*4 + ((kk>>1)&1)*2 + (kk&1)
//  fragment {W[k][n],W[k+1][n]} (k = 4*ks + 2*khalf) = one aligned ds_load_b64 at
//  dword (ks*NOUT + n)*4 + khalf*2. Lanes 0-15 hit banks {4n,4n+1}, lanes 16-31
//  {4n+2,4n+3}: all 64 banks exactly once.
//
// All NT B fragments of a k-step are loaded into registers BEFORE the WMMA burst so
// the scheduler can batch the ds_loads and drain them with staged s_wait_dscnt.
template <int NOUT, bool RELU>
__global__ __launch_bounds__(256)
void sage_gemm_kernel(const float* __restrict__ h,
                      const float* __restrict__ neigh,
                      const float* __restrict__ deg_inv,
                      const float* __restrict__ w_self,
                      const float* __restrict__ w_neigh,
                      const float* __restrict__ bias,
                      float* __restrict__ out,
                      int n_nodes) {
  __shared__ float lds_w[DFEAT * NOUT];  // <= 64 KB

  const int tid   = threadIdx.x;
  const int wave  = tid >> 5;
  const int lane  = tid & 31;
  const int khalf = lane >> 4;    // 0: lanes 0-15, 1: lanes 16-31
  const int lrow  = lane & 15;
  const int m_base = blockIdx.x * 128 + wave * 16;
  constexpr int NT = NOUT / 16;

  const int m  = m_base + lrow;
  const int mc = (m < n_nodes) ? m : (n_nodes - 1);  // clamp tail rows (stores masked)
  const float dinv = deg_inv[mc];

  v8f acc[NT] = {};

  // ---- phase 0: h @ w_self ----
  {
    #pragma unroll 4
    for (int i = tid; i < DFEAT * NOUT; i += 256) {
      const int kk = i / NOUT, n = i % NOUT;
      lds_w[((kk >> 2) * NOUT + n) * 4 + ((kk >> 1) & 1) * 2 + (kk & 1)] = w_self[i];
    }
    __syncthreads();

    const float* A = h + (long)mc * DFEAT;
    #pragma unroll 4
    for (int ks = 0; ks < DFEAT / 4; ++ks) {
      const v2f a = *(const v2f*)(A + ks * 4 + 2 * khalf);
      v2f bf[NT];
      #pragma unroll
      for (int nt = 0; nt < NT; ++nt)
        bf[nt] = *(const v2f*)(lds_w + (ks * NOUT + nt * 16 + lrow) * 4 + khalf * 2);
      #pragma unroll
      for (int nt = 0; nt < NT; ++nt)
        acc[nt] = __builtin_amdgcn_wmma_f32_16x16x4_f32(
            false, a, false, bf[nt], (short)0, acc[nt], false, false);
    }
    __syncthreads();
  }

  // ---- phase 1: (neigh * deg_inv) @ w_neigh ----
  {
    #pragma unroll 4
    for (int i = tid; i < DFEAT * NOUT; i += 256) {
      const int kk = i / NOUT, n = i % NOUT;
      lds_w[((kk >> 2) * NOUT + n) * 4 + ((kk >> 1) & 1) * 2 + (kk & 1)] = w_neigh[i];
    }
    __syncthreads();

    const float* A = neigh + (long)mc * DFEAT;
    #pragma unroll 4
    for (int ks = 0; ks < DFEAT / 4; ++ks) {
      v2f a = *(const v2f*)(A + ks * 4 + 2 * khalf);
      a.x *= dinv;
      a.y *= dinv;
      v2f bf[NT];
      #pragma unroll
      for (int nt = 0; nt < NT; ++nt)
        bf[nt] = *(const v2f*)(lds_w + (ks * NOUT + nt * 16 + lrow) * 4 + khalf * 2);
      #pragma unroll
      for (int nt = 0; nt < NT; ++nt)
        acc[nt] = __builtin_amdgcn_wmma_f32_16x16x4_f32(
            false, a, false, bf[nt], (short)0, acc[nt], false, false);
    }
    __syncthreads();
  }

  // ---- epilogue: bias (+ ReLU), masked store ----
  #pragma unroll
  for (int nt = 0; nt < NT; ++nt) {
    const int n = nt * 16 + lrow;
    const float bv = bias[n];
    #pragma unroll
    for (int j = 0; j < 8; ++j) {
      const int mrow = m_base + j + 8 * khalf;
      if (mrow < n_nodes) {
        float v = acc[nt][j] + bv;
        if (RELU) v = (v > 0.0f) ? v : 0.0f;
        out[(long)mrow * NOUT + n] = v;
      }
    }
  }
}

// ---------------------------------------------------------------- launcher
extern "C" void kernel_launch(void* const* d_in, const int* in_sizes, int n_in,
                              void* d_out, int out_size, void* d_ws, size_t ws_size,
                              hipStream_t stream) {
  (void)in_sizes; (void)n_in; (void)out_size; (void)ws_size;

  const float* x        = (const float*)d_in[0];
  const int*   src      = (const int*)  d_in[1];
  const int*   dst      = (const int*)  d_in[2];
  const float* w_self0  = (const float*)d_in[3];
  const float* w_neigh0 = (const float*)d_in[4];
  const float* b0       = (const float*)d_in[5];
  const float* w_self1  = (const float*)d_in[6];
  const float* w_neigh1 = (const float*)d_in[7];
  const float* b1       = (const float*)d_in[8];
  const float* w_self2  = (const float*)d_in[9];
  const float* w_neigh2 = (const float*)d_in[10];
  const float* b2       = (const float*)d_in[11];
  float* out = (float*)d_out;

  // workspace layout (f32): deg_inv[N] | neigh[N*128] | hA[N*128] | hB[N*128]  (~154 MB)
  float* deg   = (float*)d_ws;
  float* neigh = deg   + NNODES;
  float* hA    = neigh + (size_t)NNODES * DFEAT;
  float* hB    = hA    + (size_t)NNODES * DFEAT;

  const long feat4 = (long)NNODES * DFEAT / 4;
  const int  gemm_blocks = (NNODES + 127) / 128;       // 782
  const int  scat_blocks = (NEDGES + 7) / 8;           // 8 waves/block, 1 wave/edge

  // degree -> deg_inv (in place)
  zero4_kernel<<<256, 256, 0, stream>>>((float4*)deg, NNODES / 4);
  degree_kernel<<<(NEDGES + 255) / 256, 256, 0, stream>>>(dst, deg, NEDGES);
  deginv_kernel<<<(NNODES + 255) / 256, 256, 0, stream>>>(deg, NNODES);

  // layer 0: x -> hA (ReLU)
  zero4_kernel<<<4096, 256, 0, stream>>>((float4*)neigh, feat4);
  scatter_kernel<<<scat_blocks, 256, 0, stream>>>(x, src, dst, neigh, NEDGES);
  sage_gemm_kernel<128, true><<<gemm_blocks, 256, 0, stream>>>(
      x, neigh, deg, w_self0, w_neigh0, b0, hA, NNODES);

  // layer 1: hA -> hB (ReLU)
  zero4_kernel<<<4096, 256, 0, stream>>>((float4*)neigh, feat4);
  scatter_kernel<<<scat_blocks, 256, 0, stream>>>(hA, src, dst, neigh, NEDGES);
  sage_gemm_kernel<128, true><<<gemm_blocks, 256, 0, stream>>>(
      hA, neigh, deg, w_self1, w_neigh1, b1, hB, NNODES);

  // layer 2: hB -> out (no ReLU), NOUT=64
  zero4_kernel<<<4096, 256, 0, stream>>>((float4*)neigh, feat4);
  scatter_kernel<<<scat_blocks, 256, 0, stream>>>(hB, src, dst, neigh, NEDGES);
  sage_gemm_kernel<64, false><<<gemm_blocks, 256, 0, stream>>>(
      hB, neigh, deg, w_self2, w_neigh2, b2, out, NNODES);
}